// LightGCN_63153199120971
// MI455X (gfx1250) — compile-verified
//
#include <hip/hip_runtime.h>

#define DIM 64
#define REG_W 1e-4f

typedef float v2f __attribute__((ext_vector_type(2)));
typedef float v8f __attribute__((ext_vector_type(8)));

// ---------------- utility: zero a float range ----------------
__global__ void zero_f32(float* __restrict__ p, long long n) {
  long long i = (long long)blockIdx.x * blockDim.x + threadIdx.x;
  long long s = (long long)gridDim.x * blockDim.x;
  for (; i < n; i += s) p[i] = 0.0f;
}

// ---------------- x0 = [user_table ; item_table] (float4 vectorized) ----------------
__global__ void concat_tables(const float4* __restrict__ ut, const float4* __restrict__ it,
                              float4* __restrict__ x, long long nu4, long long tot4) {
  long long i = (long long)blockIdx.x * blockDim.x + threadIdx.x;
  long long s = (long long)gridDim.x * blockDim.x;
  for (; i < tot4; i += s) x[i] = (i < nu4) ? ut[i] : it[i - nu4];
}

// ---------------- degree histogram (exact: integer-valued f32 adds) ----------------
__global__ void degree_kernel(const int* __restrict__ eu, const int* __restrict__ ei,
                              float* __restrict__ du, float* __restrict__ di, int E) {
  int i = blockIdx.x * blockDim.x + threadIdx.x;
  int s = gridDim.x * blockDim.x;
  for (; i < E; i += s) {
    atomicAdd(&du[eu[i]], 1.0f);
    atomicAdd(&di[ei[i]], 1.0f);
  }
}

// ---------------- w[e] = rsqrt(max(deg_u*deg_i, 1)) ----------------
__global__ void weight_kernel(const int* __restrict__ eu, const int* __restrict__ ei,
                              const float* __restrict__ du, const float* __restrict__ di,
                              float* __restrict__ w, int E) {
  int i = blockIdx.x * blockDim.x + threadIdx.x;
  int s = gridDim.x * blockDim.x;
  for (; i < E; i += s) {
    float d = du[eu[i]] * di[ei[i]];
    w[i] = 1.0f / sqrtf(fmaxf(d, 1.0f));
  }
}

// ---------------- one wave per edge: y[u] += w*x[i+nu]; y[i+nu] += w*x[u] ----------------
// 32 lanes * float2 = full 64-float row; coalesced 256B loads, f32 L2 atomics.
__global__ void spmm_scatter(const int* __restrict__ eu, const int* __restrict__ ei,
                             const float* __restrict__ w, const float* __restrict__ x,
                             float* __restrict__ y, int E, int nu) {
  int lane = threadIdx.x & 31;
  long long wid = ((long long)blockIdx.x * blockDim.x + threadIdx.x) >> 5;
  long long nw  = ((long long)gridDim.x * blockDim.x) >> 5;
  for (long long e = wid; e < E; e += nw) {
    int u = eu[e];
    int v = ei[e] + nu;
    float we = w[e];
    v2f a = *(const v2f*)(x + (long long)u * DIM + 2 * lane);
    v2f b = *(const v2f*)(x + (long long)v * DIM + 2 * lane);
    float* yu = y + (long long)u * DIM + 2 * lane;
    float* yv = y + (long long)v * DIM + 2 * lane;
    atomicAdd(yv,     we * a.x);
    atomicAdd(yv + 1, we * a.y);
    atomicAdd(yu,     we * b.x);
    atomicAdd(yu + 1, we * b.y);
  }
}

// ---------------- acc[b,:] += x[idx[b]+base, :] (layer-sum at gathered rows only) ----------------
__global__ void gather_acc(const float* __restrict__ x, const int* __restrict__ idx,
                           float* __restrict__ acc, int B, int base) {
  long long n = (long long)B * DIM;
  long long i = (long long)blockIdx.x * blockDim.x + threadIdx.x;
  long long s = (long long)gridDim.x * blockDim.x;
  for (; i < n; i += s) {
    int b = (int)(i >> 6);
    int k = (int)(i & 63);
    acc[i] += x[((long long)idx[b] + base) * DIM + k];
  }
}

// ---------------- WMMA f32 16x16x4 batched-dot scoring ----------------
// One wave = 16 batch rows. A[m][k] = su[row][k]; B[k][n] = sp/sn[row_n][k].
// Chain 16 WMMAs over K=64, then read the C diagonal per the 16x16 f32 layout.
__global__ void wmma_scores(const float* __restrict__ su, const float* __restrict__ sp,
                            const float* __restrict__ sn, float* __restrict__ outp,
                            float* __restrict__ outn, int B) {
  int lane = threadIdx.x & 31;
  int tile = (int)(((long long)blockIdx.x * blockDim.x + threadIdx.x) >> 5);
  int ntiles = B >> 4;
  if (tile >= ntiles) return;              // wave-uniform: EXEC stays all-ones
  int m = lane & 15;
  long long row = (long long)tile * 16 + m;
  int ksel = (lane < 16) ? 0 : 2;          // A/B half-wave K split per ISA layout
  v8f cp = {0.f,0.f,0.f,0.f,0.f,0.f,0.f,0.f};
  v8f cn = {0.f,0.f,0.f,0.f,0.f,0.f,0.f,0.f};
#pragma unroll
  for (int kc = 0; kc < 16; ++kc) {
    int k = kc * 4 + ksel;
    v2f a  = *(const v2f*)(su + row * DIM + k);
    v2f bp = *(const v2f*)(sp + row * DIM + k);
    v2f bn = *(const v2f*)(sn + row * DIM + k);
    cp = __builtin_amdgcn_wmma_f32_16x16x4_f32(false, a, false, bp, (short)0, cp, false, false);
    cn = __builtin_amdgcn_wmma_f32_16x16x4_f32(false, a, false, bn, (short)0, cn, false, false);
  }
  // Diagonal (j,j): j<8 -> lane j, VGPR j ; j>=8 -> lane j+16, VGPR j-8
  int r = (lane < 8) ? lane : ((lane >= 24) ? (lane - 24) : -1);
  float dp = cp[0], dn = cn[0];
#pragma unroll
  for (int t = 1; t < 8; ++t) {
    if (r == t) { dp = cp[t]; dn = cn[t]; }
  }
  if (r >= 0) {
    int j = (lane < 8) ? lane : (lane - 16);
    // acc/4 on both operands -> divide dot by 16
    outp[tile * 16 + j] = dp * 0.0625f;
    outn[tile * 16 + j] = dn * 0.0625f;
  }
}

// ---------------- reg loss: REG_W/B * sum of squared layer-0 gathered rows ----------------
__global__ void reg_kernel(const float* __restrict__ ut, const float* __restrict__ it,
                           const int* __restrict__ users, const int* __restrict__ pos,
                           const int* __restrict__ neg, float* __restrict__ out,
                           int B, float scale) {
  long long BD = (long long)B * DIM;
  long long n = 3 * BD;
  long long i = (long long)blockIdx.x * blockDim.x + threadIdx.x;
  long long st = (long long)gridDim.x * blockDim.x;
  float s = 0.0f;
  for (; i < n; i += st) {
    int which = (int)(i / BD);
    long long rem = i - (long long)which * BD;
    int b = (int)(rem >> 6);
    int k = (int)(rem & 63);
    float v;
    if (which == 0)      v = ut[(long long)users[b] * DIM + k];
    else if (which == 1) v = it[(long long)pos[b] * DIM + k];
    else                 v = it[(long long)neg[b] * DIM + k];
    s += v * v;
  }
  for (int o = 16; o > 0; o >>= 1) s += __shfl_xor(s, o, 32);
  if ((threadIdx.x & 31) == 0) atomicAdd(out, s * scale);
}

static inline unsigned grid_for(long long n, int t, unsigned cap = 16384u) {
  long long b = (n + t - 1) / t;
  if (b < 1) b = 1;
  if (b > (long long)cap) b = cap;
  return (unsigned)b;
}

extern "C" void kernel_launch(void* const* d_in, const int* in_sizes, int n_in,
                              void* d_out, int out_size, void* d_ws, size_t ws_size,
                              hipStream_t stream) {
  (void)n_in; (void)ws_size;
  const float* user_table = (const float*)d_in[0];
  const float* item_table = (const float*)d_in[1];
  const int* edge_u   = (const int*)d_in[2];
  const int* edge_i   = (const int*)d_in[3];
  const int* users    = (const int*)d_in[4];
  const int* pos_items= (const int*)d_in[5];
  const int* neg_items= (const int*)d_in[6];

  const int nu = in_sizes[0] / DIM;
  const int ni = in_sizes[1] / DIM;
  const int E  = in_sizes[2];
  const int B  = in_sizes[4];
  const long long N  = (long long)nu + ni;
  const long long Nd = N * DIM;

  // workspace layout (floats): x0 | x1 | deg_u | deg_i | w | s_u | s_p | s_n
  float* ws = (float*)d_ws;
  float* x0 = ws;
  float* x1 = x0 + Nd;
  float* du = x1 + Nd;
  float* di = du + nu;
  float* w  = di + ni;
  float* su = w + E;
  float* sp = su + (long long)B * DIM;
  float* sn = sp + (long long)B * DIM;
  float* out = (float*)d_out;

  const int T = 256;

  // init
  zero_f32<<<grid_for((long long)nu + ni, T), T, 0, stream>>>(du, (long long)nu + ni);
  zero_f32<<<grid_for(3LL * B * DIM, T), T, 0, stream>>>(su, 3LL * B * DIM);
  zero_f32<<<grid_for(out_size, T), T, 0, stream>>>(out, out_size);

  // x0 = [user_table ; item_table]
  const long long nu4 = (long long)nu * (DIM / 4), tot4 = N * (DIM / 4);
  concat_tables<<<grid_for(tot4, T), T, 0, stream>>>(
      (const float4*)user_table, (const float4*)item_table, (float4*)x0, nu4, tot4);

  // normalization weights
  degree_kernel<<<grid_for(E, T), T, 0, stream>>>(edge_u, edge_i, du, di, E);
  weight_kernel<<<grid_for(E, T), T, 0, stream>>>(edge_u, edge_i, du, di, w, E);

  // layer 0 contribution to accumulators
  gather_acc<<<grid_for((long long)B * DIM, T), T, 0, stream>>>(x0, users,     su, B, 0);
  gather_acc<<<grid_for((long long)B * DIM, T), T, 0, stream>>>(x0, pos_items, sp, B, nu);
  gather_acc<<<grid_for((long long)B * DIM, T), T, 0, stream>>>(x0, neg_items, sn, B, nu);

  // 3 propagation layers (ping-pong x0/x1)
  float* src = x0;
  float* dst = x1;
  for (int L = 0; L < 3; ++L) {
    zero_f32<<<grid_for(Nd, T), T, 0, stream>>>(dst, Nd);
    spmm_scatter<<<4096, T, 0, stream>>>(edge_u, edge_i, w, src, dst, E, nu);
    gather_acc<<<grid_for((long long)B * DIM, T), T, 0, stream>>>(dst, users,     su, B, 0);
    gather_acc<<<grid_for((long long)B * DIM, T), T, 0, stream>>>(dst, pos_items, sp, B, nu);
    gather_acc<<<grid_for((long long)B * DIM, T), T, 0, stream>>>(dst, neg_items, sn, B, nu);
    float* tmp = src; src = dst; dst = tmp;
  }

  // WMMA scoring: exactly B/16 waves -> EXEC all-ones
  const int ntiles = B / 16;
  const int sthreads = ntiles * 32;
  wmma_scores<<<(sthreads + T - 1) / T, T, 0, stream>>>(su, sp, sn, out, out + B, B);

  // regularization scalar
  reg_kernel<<<grid_for(3LL * B * DIM, T), T, 0, stream>>>(
      user_table, item_table, users, pos_items, neg_items, out + 2 * B, B,
      REG_W / (float)B);
}